// TreeProbe_20177756356731
// MI455X (gfx1250) — compile-verified
//
#include <hip/hip_runtime.h>

typedef float v2f __attribute__((ext_vector_type(2)));
typedef float v8f __attribute__((ext_vector_type(8)));

#define BATCH 512
#define S     66
#define SP    80      // S padded to multiple of 16
#define ND    304     // N_DIM (19 * 16)
#define BS    128     // B_SIZE (8 * 16)
#define KC    76      // K chunk = 19 wmma k-steps (= 19 float4 per row)
#define KQ    19      // KC / 4
#define NCHUNK 4      // 4 * 76 = 304

#define XR 84         // xs row stride: 84%64=20, gcd(20,64)=4 -> 16 distinct banks
                      // across 16 lanes for A-frag reads; multiple of 4 -> b128 staging
#define WR 129        // wt row stride (bank stride 1 for B-frag reads)
#define TR 129        // t  row stride

__global__ __launch_bounds__(256)
void treeprobe_pairdist_kernel(const float* __restrict__ x,
                               const float* __restrict__ W,
                               const float* __restrict__ bias,
                               float* __restrict__ out)
{
    __shared__ float xs[SP * XR];   // x row-block chunk   (26,880 B)
    __shared__ float wt[KC * WR];   // W^T chunk           (39,216 B)
    __shared__ float tt[SP * TR];   // t = xW^T + b        (41,280 B)
    __shared__ float nrm[SP];       // row squared norms

    const int b     = blockIdx.x;
    const int tid   = threadIdx.x;
    const int lane  = tid & 31;
    const int wave  = tid >> 5;
    const int l15   = lane & 15;
    const int khalf = (lane >> 4) << 1;   // 0 for lanes 0-15, 2 for lanes 16-31

    const float* xb = x + (size_t)b * S * ND;

    // ---- Stage 1: t = x * W^T, 40 tiles; wave w owns n-tile w, m-tiles 0..4 ----
    v8f acc[5];
    #pragma unroll
    for (int i = 0; i < 5; ++i) acc[i] = (v8f){0.f,0.f,0.f,0.f,0.f,0.f,0.f,0.f};

    for (int c = 0; c < NCHUNK; ++c) {
        const int kc = c * KC;
        __syncthreads();   // protect previous chunk's LDS readers

        // stage x chunk: coalesced b128 global reads -> b128 LDS stores
        for (int idx = tid; idx < S * KQ; idx += 256) {
            int r = idx / KQ, q = idx - r * KQ;
            *(float4*)&xs[r * XR + 4 * q] =
                *(const float4*)(xb + (size_t)r * ND + kc + 4 * q);
        }
        // zero pad rows 66..79
        for (int idx = tid; idx < (SP - S) * KQ; idx += 256) {
            int r = S + idx / KQ, q = idx % KQ;
            *(float4*)&xs[r * XR + 4 * q] = make_float4(0.f, 0.f, 0.f, 0.f);
        }
        // stage W chunk transposed: wt[k][n] = W[n][kc+k]
        // (b128 coalesced global reads; LDS write stride 129 dwords -> bank stride 1)
        for (int idx = tid; idx < BS * KQ; idx += 256) {
            int n = idx / KQ, q = idx - n * KQ;
            float4 w4 = *(const float4*)(W + (size_t)n * ND + kc + 4 * q);
            wt[(4 * q + 0) * WR + n] = w4.x;
            wt[(4 * q + 1) * WR + n] = w4.y;
            wt[(4 * q + 2) * WR + n] = w4.z;
            wt[(4 * q + 3) * WR + n] = w4.w;
        }

        // prefetch next x chunk into cache while this chunk's WMMAs run
        if (c + 1 < NCHUNK) {
            const float* nx = xb + kc + KC;
            for (int idx = tid; idx < S * 3; idx += 256) {
                int r = idx / 3, seg = idx - r * 3;       // 3 x 128B covers 304B row
                __builtin_prefetch(nx + (size_t)r * ND + seg * 32, 0, 0);
            }
        }
        __syncthreads();

        #pragma unroll
        for (int i = 0; i < 5; ++i) {
            const int tile = wave + i * 8;
            const int mt = tile >> 3, nt = tile & 7;
            const int arow = (mt * 16 + l15) * XR;
            const int bcol = nt * 16 + l15;
            v8f cacc = acc[i];
            for (int s = 0; s < KC; s += 4) {
                const int ko = s + khalf;
                v2f a, bb;
                a.x  = xs[arow + ko];
                a.y  = xs[arow + ko + 1];
                bb.x = wt[ko * WR + bcol];
                bb.y = wt[(ko + 1) * WR + bcol];
                cacc = __builtin_amdgcn_wmma_f32_16x16x4_f32(
                         false, a, false, bb, (short)0, cacc, false, false);
            }
            acc[i] = cacc;
        }
    }

    // ---- write t (+bias) to LDS ----
    __syncthreads();
    #pragma unroll
    for (int i = 0; i < 5; ++i) {
        const int tile = wave + i * 8;
        const int mt = tile >> 3, nt = tile & 7;
        const int col   = nt * 16 + l15;
        const float bv  = bias[col];
        const int rbase = mt * 16 + (lane >> 4) * 8;
        v8f cacc = acc[i];
        #pragma unroll
        for (int r = 0; r < 8; ++r)
            tt[(rbase + r) * TR + col] = cacc[r] + bv;
    }
    __syncthreads();

    // ---- row squared norms ----
    if (tid < SP) {
        float s = 0.0f;
        const float* row = &tt[tid * TR];
        for (int k = 0; k < BS; ++k) s += row[k] * row[k];
        nrm[tid] = s;
    }
    __syncthreads();

    // ---- Stage 2: G = t * t^T (25 tiles), out = nrm_i + nrm_j - 2 G_ij ----
    float* outb = out + (size_t)b * S * S;
    for (int tile = wave; tile < 25; tile += 8) {
        const int it = tile / 5, jt = tile - it * 5;
        const int arow = (it * 16 + l15) * TR;
        const int brow = (jt * 16 + l15) * TR;   // B = t^T: same fragment pattern
        v8f g = (v8f){0.f,0.f,0.f,0.f,0.f,0.f,0.f,0.f};
        for (int k = 0; k < BS; k += 4) {
            const int ko = k + khalf;
            v2f a, bb;
            a.x  = tt[arow + ko];
            a.y  = tt[arow + ko + 1];
            bb.x = tt[brow + ko];
            bb.y = tt[brow + ko + 1];
            g = __builtin_amdgcn_wmma_f32_16x16x4_f32(
                  false, a, false, bb, (short)0, g, false, false);
        }
        const int gj = jt * 16 + l15;
        if (gj < S) {
            const float nj = nrm[gj];
            const int gibase = it * 16 + (lane >> 4) * 8;
            #pragma unroll
            for (int r = 0; r < 8; ++r) {
                const int gi = gibase + r;
                if (gi < S)
                    outb[(size_t)gi * S + gj] = nrm[gi] + nj - 2.0f * g[r];
            }
        }
    }
}

extern "C" void kernel_launch(void* const* d_in, const int* in_sizes, int n_in,
                              void* d_out, int out_size, void* d_ws, size_t ws_size,
                              hipStream_t stream) {
    const float* x    = (const float*)d_in[0];
    const float* W    = (const float*)d_in[1];
    const float* bias = (const float*)d_in[2];
    float* out        = (float*)d_out;
    hipLaunchKernelGGL(treeprobe_pairdist_kernel,
                       dim3(BATCH), dim3(256), 0, stream,
                       x, W, bias, out);
}